// VariationalLinearEncoder_85907935854601
// MI455X (gfx1250) — compile-verified
//
#include <hip/hip_runtime.h>
#include <hip/hip_bf16.h>

typedef __attribute__((ext_vector_type(2))) float v2f;
typedef __attribute__((ext_vector_type(8))) float v8f;

#define IN_CH   256
#define OUT_CH  10
#define HCOLS   32   // padded combined output columns: [0..9]=mu, [16..25]=logstd

// ---------------------------------------------------------------------------
// Kernel 1: pack W_mu / W_logstd into a K-major padded weight array.
// WpT[j*256 + k] = combined W[k][j]; so a float2 load at (j*256+k) gives
// W[k][j], W[k+1][j] -> exactly the B-fragment layout for V_WMMA_F32_16X16X4_F32.
// ---------------------------------------------------------------------------
__global__ void pack_weights_kernel(const float* __restrict__ Wmu,
                                    const float* __restrict__ Wls,
                                    float* __restrict__ WpT) {
    int idx = blockIdx.x * blockDim.x + threadIdx.x;
    if (idx >= HCOLS * IN_CH) return;
    int j = idx / IN_CH;
    int k = idx - j * IN_CH;
    float v = 0.0f;
    if (j < OUT_CH)                       v = Wmu[k * OUT_CH + j];
    else if (j >= 16 && j < 16 + OUT_CH)  v = Wls[k * OUT_CH + (j - 16)];
    WpT[j * IN_CH + k] = v;
}

// ---------------------------------------------------------------------------
// Degree / normalization kernels
// ---------------------------------------------------------------------------
__global__ void init_deg_kernel(float* __restrict__ deg, int n) {
    int i = blockIdx.x * blockDim.x + threadIdx.x;
    if (i < n) deg[i] = 1.0f;           // self-loop contributes 1
}

__global__ void edge_deg_kernel(const long long* __restrict__ ei,
                                float* __restrict__ deg, int nEdges) {
    int e = blockIdx.x * blockDim.x + threadIdx.x;
    if (e >= nEdges) return;
    long long c = ei[(size_t)nEdges + e];   // target node
    unsafeAtomicAdd(&deg[c], 1.0f);         // -> global_atomic_add_f32
}

__global__ void dinv_kernel(float* __restrict__ deg, int n) {
    int i = blockIdx.x * blockDim.x + threadIdx.x;
    if (i < n) deg[i] = rsqrtf(fmaxf(deg[i], 1.0f));
}

// ---------------------------------------------------------------------------
// Kernel 2: h[N x 32] = x[N x 256] @ Wp[256 x 32] via V_WMMA_F32_16X16X4_F32.
// One wave per 16-row tile; each wave computes BOTH 16-col tiles so the A
// fragment (x) is loaded exactly once from HBM.
// A frag (16x4 f32): lane L holds row M=L&15; lanes<16: K=k,k+1; lanes>=16: K=k+2,k+3.
// B frag (4x16 f32): lane L holds col N=L&15; same K split -> float2 from WpT.
// D (16x16 f32): VGPR r holds M = r + 8*(L>=16), N = L&15.
// ---------------------------------------------------------------------------
__global__ void __launch_bounds__(256)
gemm_wmma_kernel(const float* __restrict__ x, const float* __restrict__ WpT,
                 float* __restrict__ h, int nNodes) {
    const int lane = threadIdx.x & 31;
    const int wave = threadIdx.x >> 5;
    const int tile = blockIdx.x * 8 + wave;
    const int rowBase = tile * 16;
    if (rowBase >= nNodes) return;               // wave-uniform: EXEC stays all-1s

    const int m    = lane & 15;
    const int kOff = (lane >> 4) * 2;            // 0 for lanes 0-15, 2 for 16-31

    int row  = rowBase + m;
    int rowC = (row < nNodes) ? row : (nNodes - 1);   // clamp keeps EXEC full
    const float* aPtr  = x   + (size_t)rowC * IN_CH + kOff;
    const float* b0Ptr = WpT + (size_t)m * IN_CH + kOff;          // cols 0..15
    const float* b1Ptr = WpT + (size_t)(16 + m) * IN_CH + kOff;   // cols 16..31

    v8f c0 = {};
    v8f c1 = {};
#pragma unroll 8
    for (int k = 0; k < IN_CH; k += 4) {
        v2f a  = *(const v2f*)(aPtr  + k);
        v2f b0 = *(const v2f*)(b0Ptr + k);
        v2f b1 = *(const v2f*)(b1Ptr + k);
        c0 = __builtin_amdgcn_wmma_f32_16x16x4_f32(
                 false, a, false, b0, (short)0, c0, false, false);
        c1 = __builtin_amdgcn_wmma_f32_16x16x4_f32(
                 false, a, false, b1, (short)0, c1, false, false);
    }

    const int n    = lane & 15;
    const int mOff = (lane >> 4) * 8;
#pragma unroll
    for (int r = 0; r < 8; ++r) {
        int mm = rowBase + mOff + r;
        if (mm < nNodes) {
            h[(size_t)mm * HCOLS + n]      = c0[r];
            h[(size_t)mm * HCOLS + 16 + n] = c1[r];
        }
    }
}

// ---------------------------------------------------------------------------
// Kernel 3: initialize output with bias + self-loop term:
//   out[i][j] = b[j] + dinv[i]^2 * h[i][j]
// ---------------------------------------------------------------------------
__global__ void init_out_kernel(const float* __restrict__ dinv,
                                const float* __restrict__ h,
                                const float* __restrict__ bmu,
                                const float* __restrict__ bls,
                                float* __restrict__ out, int nNodes) {
    int idx = blockIdx.x * blockDim.x + threadIdx.x;
    if (idx >= nNodes * OUT_CH) return;
    int i = idx / OUT_CH;
    int j = idx - i * OUT_CH;
    float di = dinv[i];
    float w  = di * di;
    const float* hr = h + (size_t)i * HCOLS;
    out[idx]                              = bmu[j] + w * hr[j];
    out[(size_t)nNodes * OUT_CH + idx]    = bls[j] + w * hr[16 + j];
}

// ---------------------------------------------------------------------------
// Kernel 4: edge scatter: out[c] += dinv[r]*dinv[c] * h[r]  (mu and logstd).
// h (12.8 MB) and out (8 MB) are L2-resident; atomics are HW f32 adds.
// ---------------------------------------------------------------------------
__global__ void scatter_kernel(const long long* __restrict__ ei,
                               const float* __restrict__ dinv,
                               const float* __restrict__ h,
                               float* __restrict__ out, int nEdges, int nNodes) {
    int e = blockIdx.x * blockDim.x + threadIdx.x;
    if (e >= nEdges) return;
    long long r = ei[e];                      // source
    long long c = ei[(size_t)nEdges + e];     // target
    float w = dinv[r] * dinv[c];
    const float* hr = h + (size_t)r * HCOLS;
    float* mu = out + (size_t)c * OUT_CH;
    float* ls = out + (size_t)nNodes * OUT_CH + (size_t)c * OUT_CH;
#pragma unroll
    for (int j = 0; j < OUT_CH; ++j) {
        unsafeAtomicAdd(&mu[j], w * hr[j]);
        unsafeAtomicAdd(&ls[j], w * hr[16 + j]);
    }
}

// ---------------------------------------------------------------------------
extern "C" void kernel_launch(void* const* d_in, const int* in_sizes, int n_in,
                              void* d_out, int out_size, void* d_ws, size_t ws_size,
                              hipStream_t stream) {
    const float*     x   = (const float*)d_in[0];
    const long long* ei  = (const long long*)d_in[1];   // int64 edge_index [2, E]
    const float*     Wmu = (const float*)d_in[2];
    const float*     bmu = (const float*)d_in[3];
    const float*     Wls = (const float*)d_in[4];
    const float*     bls = (const float*)d_in[5];
    float*           out = (float*)d_out;

    const int nNodes = in_sizes[0] / IN_CH;   // 100000
    const int nEdges = in_sizes[1] / 2;       // 3200000

    // Workspace layout (floats): WpT[32*256] | dinv[nNodes] | h[nNodes*32]
    float* ws   = (float*)d_ws;
    float* WpT  = ws;
    float* dinv = ws + (size_t)HCOLS * IN_CH;
    size_t hOff = (size_t)HCOLS * IN_CH + (((size_t)nNodes + 127) / 128) * 128;
    float* h    = ws + hOff;

    const int T = 256;
    pack_weights_kernel<<<(HCOLS * IN_CH + T - 1) / T, T, 0, stream>>>(Wmu, Wls, WpT);
    init_deg_kernel<<<(nNodes + T - 1) / T, T, 0, stream>>>(dinv, nNodes);
    edge_deg_kernel<<<(nEdges + T - 1) / T, T, 0, stream>>>(ei, dinv, nEdges);
    dinv_kernel<<<(nNodes + T - 1) / T, T, 0, stream>>>(dinv, nNodes);

    int tiles  = (nNodes + 15) / 16;          // 16-row tiles, 1 wave each
    int blocks = (tiles + 7) / 8;             // 8 waves / 256-thread block
    gemm_wmma_kernel<<<blocks, T, 0, stream>>>(x, WpT, h, nNodes);

    init_out_kernel<<<(nNodes * OUT_CH + T - 1) / T, T, 0, stream>>>(
        dinv, h, bmu, bls, out, nNodes);
    scatter_kernel<<<(nEdges + T - 1) / T, T, 0, stream>>>(
        ei, dinv, h, out, nEdges, nNodes);
}